// GMTransposePool_2147483648731
// MI455X (gfx1250) — compile-verified
//
#include <hip/hip_runtime.h>
#include <stdint.h>

typedef __attribute__((ext_vector_type(2))) float v2f;
typedef __attribute__((ext_vector_type(8))) float v8f;

#define B_DIM   64
#define CIN_D   64
#define K_D     4096
#define COUT_D  64
#define F_D     4
#define L_D     (F_D * K_D)            // 16384
#define OTILE   4                      // output channels per workgroup
#define ROWS    (OTILE * F_D)          // 16 = WMMA M
#define NWAVES  8
#define BLOCK   256
#define RES_FLOATS (ROWS * K_D)        // 65536 floats = 256 KB
#define IDX_BASE_B (RES_FLOATS * 4)    // idx region starts at 256 KB
#define LDS_BYTES  (IDX_BASE_B + L_D * 4)   // 320 KB total

__global__ __launch_bounds__(BLOCK)
void upsample_gather_fused(const float* __restrict__ x,
                           const float* __restrict__ w,
                           const int*   __restrict__ idx,
                           float*       __restrict__ out)
{
    extern __shared__ float lds[];          // [0,256K): res slab; [256K,320K): indices

    const int b    = blockIdx.y;            // batch
    const int ot   = blockIdx.x;            // o-tile; o0 = ot*OTILE
    const int tid  = threadIdx.x;
    const int wave = tid >> 5;
    const int lane = tid & 31;
    const int n    = lane & 15;             // column-in-tile / row m for A loads
    const int half = lane >> 4;             // fp32 WMMA: lanes 16-31 hold K 2..3
    const int ck0  = half * 2;

    const float* __restrict__ Xb = x + (size_t)b * CIN_D * K_D;

    // ---------- Kick off async copy of indices (64 KB) into LDS ----------
    // Overlaps with the whole GEMM phase. 16 rounds x 256 lanes x 16 B = 64 KB.
    {
        const uint64_t gbase = (uint64_t)idx;
        #pragma unroll
        for (int j = 0; j < 16; ++j) {
            const unsigned off = (unsigned)((j * BLOCK + tid) * 16);
            const unsigned dst = (unsigned)IDX_BASE_B + off;
            asm volatile("global_load_async_to_lds_b128 %0, %1, %2"
                         :: "v"(dst), "v"(off), "s"(gbase) : "memory");
        }
    }

    // ---------------- Phase 1: A (weights) into registers ----------------
    // A[m, i] = W[i, o0 + m/4, m%4] = w[i*256 + ot*16 + m]  (contiguous in m)
    v2f a[16];
    {
        const int col = ot * ROWS + n;      // m == n for this lane's A rows
        #pragma unroll
        for (int kk = 0; kk < 16; ++kk) {
            a[kk].x = w[(4 * kk + ck0    ) * (COUT_D * F_D) + col];
            a[kk].y = w[(4 * kk + ck0 + 1) * (COUT_D * F_D) + col];
        }
    }

    // ---------------- Phase 1: GEMM res slab into LDS ----------------
    // 256 column tiles of 16; 8 waves -> 32 tiles each, interleaved.
    // B fragments staged into a register array first so all 32 loads are in
    // flight before the WMMA chain starts (single-wave ILP covers L2 latency).
    for (int t = 0; t < 32; ++t) {
        const int k0 = (t * NWAVES + wave) * 16;

        v2f bv[16];
        #pragma unroll
        for (int kk = 0; kk < 16; ++kk) {
            bv[kk].x = Xb[(size_t)(4 * kk + ck0    ) * K_D + k0 + n];
            bv[kk].y = Xb[(size_t)(4 * kk + ck0 + 1) * K_D + k0 + n];
        }

        v8f c = {};
        #pragma unroll
        for (int kk = 0; kk < 16; ++kk) {
            // D = A x B + C ; fp32 16x16x4
            c = __builtin_amdgcn_wmma_f32_16x16x4_f32(
                    false, a[kk], false, bv[kk], (short)0, c, false, false);
        }

        // D layout: VGPR v -> M = v + half*8, N = n. LDS row m covers (o_local, f).
        #pragma unroll
        for (int v = 0; v < 8; ++v) {
            const int m = v + half * 8;
            lds[m * K_D + k0 + n] = c[v];
        }
    }

    // Each wave: ensure its async idx loads landed, then group barrier
    // (also orders the ds_stores via __syncthreads' dscnt wait).
    asm volatile("s_wait_asynccnt 0x0" ::: "memory");
    __syncthreads();

    // ---------------- Phase 2: gather via async LDS->global DMA ----------------
    // out[b, o0+oo, l] = lds[oo*L_D + g],  g = idx[l].
    // Lane-consecutive l => each async store writes 128 B contiguous globally.
    const int* idx_lds = (const int*)&lds[RES_FLOATS];
    const uint64_t obase =
        (uint64_t)(out + ((size_t)b * COUT_D + (size_t)ot * OTILE) * L_D);

    for (int j = 0; j < L_D / BLOCK; ++j) {       // 64 iterations
        const int l = j * BLOCK + tid;
        const int g = idx_lds[l];                 // ds_load_b32, conflict-free
        const unsigned voff = (unsigned)(l * 4);
        #pragma unroll
        for (int oo = 0; oo < OTILE; ++oo) {
            const unsigned src = (unsigned)((oo * L_D + g) * 4);
            const uint64_t ob  = obase + (uint64_t)(oo * L_D) * 4u;
            asm volatile("global_store_async_from_lds_b32 %0, %1, %2"
                         :: "v"(voff), "v"(src), "s"(ob) : "memory");
        }
    }

    // Drain async stores before wave exit.
    asm volatile("s_wait_asynccnt 0x0" ::: "memory");
}

extern "C" void kernel_launch(void* const* d_in, const int* in_sizes, int n_in,
                              void* d_out, int out_size, void* d_ws, size_t ws_size,
                              hipStream_t stream) {
    (void)in_sizes; (void)n_in; (void)d_ws; (void)ws_size; (void)out_size;
    const float* x   = (const float*)d_in[0];   // [64, 64, 1, 4096]
    const float* w   = (const float*)d_in[1];   // [64, 64, 4]
    const int*   idx = (const int*)d_in[2];     // [16384]
    float*       out = (float*)d_out;           // [64, 64, 1, 16384]

    dim3 grid(COUT_D / OTILE, B_DIM);           // 16 x 64 = 1024 workgroups
    dim3 block(BLOCK);
    upsample_gather_fused<<<grid, block, (size_t)LDS_BYTES, stream>>>(x, w, idx, out);
}